// ModelBLA_20177756356801
// MI455X (gfx1250) — compile-verified
//
#include <hip/hip_runtime.h>

// Problem constants (from reference): N=16384, NU=2, R=256, NX=8, NY=2
#define NN   16384
#define RR   256
#define LCH  128              // chunk length
#define CCH  (NN / LCH)       // 128 chunks

typedef __attribute__((ext_vector_type(2))) float v2f;
typedef __attribute__((ext_vector_type(8))) float v8f;

// VALU cross-half swap (lane i <-> lane i^16). v_permlanex16_b32 with identity
// nibble selects; falls back to ds_bpermute via __shfl_xor if unavailable.
__device__ __forceinline__ float swap16(float v) {
#if __has_builtin(__builtin_amdgcn_permlanex16)
  int i = __float_as_int(v);
  i = __builtin_amdgcn_permlanex16(i, i, 0x76543210, 0xfedcba98, false, false);
  return __int_as_float(i);
#else
  return __shfl_xor(v, 16);
#endif
}

// Element (row, col) of the 16x12 double-step matrix
//   M = [[ A,   B,  0 ],
//        [ A^2, AB, B ]]
__device__ __forceinline__ float mval(const float* __restrict__ A,
                                      const float* __restrict__ B,
                                      int row, int col) {
  if (row < 8) {
    if (col < 8)  return A[row * 8 + col];
    if (col < 10) return B[row * 2 + (col - 8)];
    return 0.0f;
  }
  const int r = row - 8;
  if (col < 8) {
    float s = 0.0f;
#pragma unroll
    for (int k = 0; k < 8; ++k) s += A[r * 8 + k] * A[k * 8 + col];
    return s;
  }
  if (col < 10) {
    float s = 0.0f;
#pragma unroll
    for (int k = 0; k < 8; ++k) s += A[r * 8 + k] * B[k * 2 + (col - 8)];
    return s;
  }
  return B[r * 2 + (col - 10)];
}

// Phases 1 and 3 share the WMMA double-step inner loop.
// grid = (CCH, RR/16), block = 32 (one wave; WMMA is whole-wave).
template <bool EMIT>
__global__ __launch_bounds__(32) void lss_chunk_kernel(
    const float* __restrict__ u,    // (N, 2, 256)
    const float* __restrict__ A,    // (8, 8)
    const float* __restrict__ B,    // (8, 2)
    const float* __restrict__ Cy,   // (2, 8)
    const float* __restrict__ Dyu,  // (2, 2)
    const float* __restrict__ um, const float* __restrict__ us,
    const float* __restrict__ ym, const float* __restrict__ ys,
    const float* __restrict__ xs_g, // (CCH, 8, 256) chunk-start states (EMIT only)
    float* __restrict__ out_y,      // (N, 2, 256)            (EMIT only)
    float* __restrict__ out_X,      // (N, 8, 256)            (EMIT only)
    float* __restrict__ v_out)      // (CCH, 8, 256)          (!EMIT only)
{
  const int lane = threadIdx.x;
  const int lo = lane & 15;
  const int hi = lane >> 4;
  const int chunk = blockIdx.x;
  const int colbase = blockIdx.y * 16;
  const int col = colbase + lo;
  const int cs = chunk * LCH;

  // A-operand registers for M (16x12): 3 K-slices of 2 VGPRs each.
  // (vgpr p, lane) holds M[lane&15][slice*4 + p + 2*(lane>=16)]
  float m[6];
#pragma unroll
  for (int s = 0; s < 3; ++s)
#pragma unroll
    for (int p = 0; p < 2; ++p)
      m[2 * s + p] = mval(A, B, lo, s * 4 + p + 2 * hi);

  const float um0 = um[0], um1 = um[1];
  const float ius0 = 1.0f / us[0], ius1 = 1.0f / us[1];

  float cy[16], dy[4], ym0 = 0.f, ym1 = 0.f, ys0 = 0.f, ys1 = 0.f;
  if (EMIT) {
#pragma unroll
    for (int i = 0; i < 16; ++i) cy[i] = Cy[i];
#pragma unroll
    for (int i = 0; i < 4; ++i) dy[i] = Dyu[i];
    ym0 = ym[0]; ym1 = ym[1]; ys0 = ys[0]; ys1 = ys[1];
  }

  // State in B-operand layout: b[2*s+q] = x[q + 2*hi + 4*s][col]
  float b0, b1, b2, b3;
  if (EMIT) {
    const float* xsc = xs_g + chunk * (8 * RR);
    b0 = xsc[(0 + 2 * hi) * RR + col];
    b1 = xsc[(1 + 2 * hi) * RR + col];
    b2 = xsc[(4 + 2 * hi) * RR + col];
    b3 = xsc[(5 + 2 * hi) * RR + col];
    // Emit y[cs] (first step of chunk) from a plain per-lane copy of xs.
    float xv[8];
#pragma unroll
    for (int v = 0; v < 8; ++v) xv[v] = xsc[v * RR + col];
    const float w0 = (u[cs * 512 + col] - um0) * ius0;
    const float w1 = (u[cs * 512 + 256 + col] - um1) * ius1;
    float y0 = dy[0] * w0 + dy[1] * w1;
    float y1 = dy[2] * w0 + dy[3] * w1;
#pragma unroll
    for (int v = 0; v < 8; ++v) { y0 += cy[v] * xv[v]; y1 += cy[8 + v] * xv[v]; }
    y0 = ym0 + ys0 * y0;
    y1 = ym1 + ys1 * y1;
    if (lane < 16) {
      __builtin_nontemporal_store(y0, &out_y[cs * 512 + col]);
      __builtin_nontemporal_store(y1, &out_y[cs * 512 + 256 + col]);
    }
  } else {
    b0 = b1 = b2 = b3 = 0.0f;
  }

  for (int t = cs; t < cs + LCH; t += 2) {
    // u slice for WMMA: rows 8..11 of s-vector = [u_t ; u_{t+1}] normalized.
    const float u0 = (u[(t + hi) * 512 + col] - um0) * ius0;
    const float u1 = (u[(t + hi) * 512 + 256 + col] - um1) * ius1;

    if (EMIT) {
      // X[t] straight from B-layout state regs (all 32 lanes useful).
      float* Xt = out_X + (size_t)t * (8 * RR);
      __builtin_nontemporal_store(b0, &Xt[(0 + 2 * hi) * RR + col]);
      __builtin_nontemporal_store(b1, &Xt[(1 + 2 * hi) * RR + col]);
      __builtin_nontemporal_store(b2, &Xt[(4 + 2 * hi) * RR + col]);
      __builtin_nontemporal_store(b3, &Xt[(5 + 2 * hi) * RR + col]);
    }

    // D(16x16) = M(16x12) x [x_t; u_t; u_{t+1}](12x16):
    //   rows 0..7  = x_{t+1},  rows 8..15 = x_{t+2}
    v8f d = {};
    d = __builtin_amdgcn_wmma_f32_16x16x4_f32(false, (v2f){m[0], m[1]},
                                              false, (v2f){b0, b1},
                                              (short)0, d, false, false);
    d = __builtin_amdgcn_wmma_f32_16x16x4_f32(false, (v2f){m[2], m[3]},
                                              false, (v2f){b2, b3},
                                              (short)0, d, false, false);
    d = __builtin_amdgcn_wmma_f32_16x16x4_f32(false, (v2f){m[4], m[5]},
                                              false, (v2f){u0, u1},
                                              (short)0, d, false, false);

    if (EMIT) {
      // X[t+1]: rows 0..7 of D. Relayout into B-row layout (VALU permlane)
      // so all 32 lanes store, fully coalesced.
      {
        const float e0 = hi ? swap16(d[2]) : d[0];
        const float e1 = hi ? swap16(d[3]) : d[1];
        const float e2 = hi ? swap16(d[6]) : d[4];
        const float e3 = hi ? swap16(d[7]) : d[5];
        float* Xt1 = out_X + (size_t)(t + 1) * (8 * RR);
        __builtin_nontemporal_store(e0, &Xt1[(0 + 2 * hi) * RR + col]);
        __builtin_nontemporal_store(e1, &Xt1[(1 + 2 * hi) * RR + col]);
        __builtin_nontemporal_store(e2, &Xt1[(4 + 2 * hi) * RR + col]);
        __builtin_nontemporal_store(e3, &Xt1[(5 + 2 * hi) * RR + col]);
      }
      // y[t+1+hi]: lane half hi holds x_{t+1+hi}[v][col] in d[v].
      const int ty = t + 1 + hi;
      if (ty < cs + LCH) {
        const float w0 = (u[ty * 512 + col] - um0) * ius0;
        const float w1 = (u[ty * 512 + 256 + col] - um1) * ius1;
        float y0 = dy[0] * w0 + dy[1] * w1;
        float y1 = dy[2] * w0 + dy[3] * w1;
#pragma unroll
        for (int v = 0; v < 8; ++v) { y0 += cy[v] * d[v]; y1 += cy[8 + v] * d[v]; }
        y0 = ym0 + ys0 * y0;
        y1 = ym1 + ys1 * y1;
        __builtin_nontemporal_store(y0, &out_y[ty * 512 + col]);
        __builtin_nontemporal_store(y1, &out_y[ty * 512 + 256 + col]);
      }
    }

    // Next-state relayout: x_{t+2} is D rows 8..15 = lanes 16..31 of d[0..7].
    // Upper-half lanes already hold their target values in-register; lower
    // half fetches them with a VALU cross-half permute.
    const float s0 = swap16(d[0]);
    const float s1 = swap16(d[1]);
    const float s4 = swap16(d[4]);
    const float s5 = swap16(d[5]);
    b0 = hi ? d[2] : s0;
    b1 = hi ? d[3] : s1;
    b2 = hi ? d[6] : s4;
    b3 = hi ? d[7] : s5;
  }

  if (!EMIT) {
    // v_c = state after L zero-init steps, in B-layout regs.
    float* vc = v_out + chunk * (8 * RR);
    vc[(0 + 2 * hi) * RR + col] = b0;
    vc[(1 + 2 * hi) * RR + col] = b1;
    vc[(4 + 2 * hi) * RR + col] = b2;
    vc[(5 + 2 * hi) * RR + col] = b3;
  }
}

// Phase 2: compute A^L by 7 squarings, then scan chunk-start states.
// grid = 8, block = 32 (32 columns per block).
__global__ __launch_bounds__(32) void lss_scan_kernel(
    const float* __restrict__ A, const float* __restrict__ x0,
    const float* __restrict__ v_in, float* __restrict__ xs_out) {
  __shared__ float P[64];
  __shared__ float T[64];
  const int tid = threadIdx.x;

  P[tid] = A[tid];
  P[tid + 32] = A[tid + 32];
  __syncthreads();
#pragma unroll
  for (int it = 0; it < 7; ++it) {  // P = A^(2^7) = A^128
#pragma unroll
    for (int e0 = 0; e0 < 2; ++e0) {
      const int e = tid + 32 * e0, i = e >> 3, j = e & 7;
      float s = 0.0f;
#pragma unroll
      for (int k = 0; k < 8; ++k) s += P[i * 8 + k] * P[k * 8 + j];
      T[e] = s;
    }
    __syncthreads();
    P[tid] = T[tid];
    P[tid + 32] = T[tid + 32];
    __syncthreads();
  }

  const int col = blockIdx.x * 32 + tid;
  float x[8];
#pragma unroll
  for (int s = 0; s < 8; ++s) x[s] = x0[s * RR + col];

  for (int c = 0; c < CCH; ++c) {
    float* xsc = xs_out + c * (8 * RR);
#pragma unroll
    for (int s = 0; s < 8; ++s) xsc[s * RR + col] = x[s];
    const float* vc = v_in + c * (8 * RR);
    float xn[8];
#pragma unroll
    for (int s = 0; s < 8; ++s) {
      float acc = vc[s * RR + col];
#pragma unroll
      for (int j = 0; j < 8; ++j) acc += P[s * 8 + j] * x[j];
      xn[s] = acc;
    }
#pragma unroll
    for (int s = 0; s < 8; ++s) x[s] = xn[s];
  }
}

extern "C" void kernel_launch(void* const* d_in, const int* in_sizes, int n_in,
                              void* d_out, int out_size, void* d_ws, size_t ws_size,
                              hipStream_t stream) {
  const float* u   = (const float*)d_in[0];
  const float* x0  = (const float*)d_in[1];
  const float* A   = (const float*)d_in[2];
  const float* Bu  = (const float*)d_in[3];
  const float* Cy  = (const float*)d_in[4];
  const float* Dyu = (const float*)d_in[5];
  const float* um  = (const float*)d_in[6];
  const float* us  = (const float*)d_in[7];
  const float* ym  = (const float*)d_in[8];
  const float* ys  = (const float*)d_in[9];

  float* out_y = (float*)d_out;                 // (N, 2, 256)
  float* out_X = out_y + (size_t)NN * 2 * RR;   // (N, 8, 256)

  float* ws_v  = (float*)d_ws;                  // (CCH, 8, 256)  ~1 MiB
  float* ws_xs = ws_v + (size_t)CCH * 8 * RR;   // (CCH, 8, 256)  ~1 MiB

  dim3 g13(CCH, RR / 16);

  // Phase 1: input-driven term per chunk (zero init, no emission).
  lss_chunk_kernel<false><<<g13, 32, 0, stream>>>(
      u, A, Bu, Cy, Dyu, um, us, ym, ys, nullptr, nullptr, nullptr, ws_v);

  // Phase 2: A^L + chunk-level scan of start states.
  lss_scan_kernel<<<8, 32, 0, stream>>>(A, x0, ws_v, ws_xs);

  // Phase 3: reconstruct all states/outputs per chunk.
  lss_chunk_kernel<true><<<g13, 32, 0, stream>>>(
      u, A, Bu, Cy, Dyu, um, us, ym, ys, ws_xs, out_y, out_X, nullptr);
}